// ASRConformer_61246233641277
// MI455X (gfx1250) — compile-verified
//
#include <hip/hip_runtime.h>
#include <hip/hip_bf16.h>

// Problem constants (from reference): B=8, T=400, U=100, V=512, BLANK=V-1
#define Bn 8
#define Tn 400
#define Un 100
#define Vn 512
#define BLANKC 511
#define NROWS (Bn * Tn * (Un + 1))   // 323200 logsumexp rows
#define NEMIT (Bn * Tn * Un)         // 320000 emit log-probs
#define NEG_INF_F (-1.0e30f)

// ---------------------------------------------------------------------------
// CDNA5 async global->LDS DMA helpers (ASYNCcnt-tracked, gfx1250-specific).
// The same signed instruction offset is added to BOTH the global address and
// the LDS address (CDNA5 ISA), so one lds-offset VGPR and
// one 64-bit global address VGPR pair cover a whole 2KB row stage.
// ---------------------------------------------------------------------------
#define ASYNC_B128_OFF(loff, gptr, IMM)                                     \
  asm volatile("global_load_async_to_lds_b128 %0, %1, off offset:" IMM      \
               :: "v"(loff), "v"(gptr) : "memory")

__device__ __forceinline__ void stage_row_async(unsigned lds_off, const float* g)
{
  // 32 lanes x 16B x 4 issues = 2048B (one V=512 f32 row)
  asm volatile("global_load_async_to_lds_b128 %0, %1, off"
               :: "v"(lds_off), "v"(g) : "memory");
  ASYNC_B128_OFF(lds_off, g, "512");
  ASYNC_B128_OFF(lds_off, g, "1024");
  ASYNC_B128_OFF(lds_off, g, "1536");
}

__device__ __forceinline__ void wait_async_le(int n)
{
  if (n == 4) asm volatile("s_wait_asynccnt 0x4" ::: "memory");
  else        asm volatile("s_wait_asynccnt 0x0" ::: "memory");
}

__device__ __forceinline__ float logaddexp_f(float a, float b)
{
  float mx = fmaxf(a, b);
  float mn = fminf(a, b);
  // exp(mn-mx) <= 1; underflows to 0 for the -1e30 sentinel -> returns mx.
  return mx + log1pf(__expf(mn - mx));
}

// ---------------------------------------------------------------------------
// Phase 1: per-row logsumexp over V=512, emit blank/token log-probs.
// One wave32 per row; rows double-buffered in LDS via async DMA.
// ---------------------------------------------------------------------------
__global__ __launch_bounds__(256) void rnnt_lse_kernel(
    const float* __restrict__ logits,   // (B,T,U+1,V)
    const int*   __restrict__ tokens,   // (B,U)
    float* __restrict__ blank_lp,       // (B,T,U+1)
    float* __restrict__ emit_lp)        // (B,T,U)
{
  __shared__ float smem[8 * 2 * Vn];    // 8 waves x double 2KB buffer = 32KB

  const int lane = threadIdx.x & 31;
  const int warp = threadIdx.x >> 5;
  const int wid  = blockIdx.x * 8 + warp;
  const int nw   = gridDim.x * 8;

  float* buf0 = smem + warp * (2 * Vn);
  float* buf1 = buf0 + Vn;
  // Low 32 bits of a generic LDS pointer == LDS byte offset (aperture is in
  // bits [63:32] only, per CDNA5 aperture check).
  const unsigned l0 = (unsigned)(uintptr_t)(void*)(buf0 + lane * 4);
  const unsigned l1 = (unsigned)(uintptr_t)(void*)(buf1 + lane * 4);

  int r = wid;
  if (r >= NROWS) return;

  stage_row_async(l0, logits + (size_t)r * Vn + lane * 4);

  int bufi = 0;
  for (;;) {
    const int  rn       = r + nw;
    const bool has_next = (rn < NROWS);
    if (has_next) {
      stage_row_async(bufi ? l0 : l1, logits + (size_t)rn * Vn + lane * 4);
      wait_async_le(4);   // next stage's 4 ops may remain outstanding
    } else {
      wait_async_le(0);
    }

    const float*  buf = bufi ? buf1 : buf0;
    const float4* bv  = (const float4*)buf;
    // Lane owns elements 4*lane + 128*k + j  (matches staged layout).
    float4 c0 = bv[lane];
    float4 c1 = bv[lane + 32];
    float4 c2 = bv[lane + 64];
    float4 c3 = bv[lane + 96];

    float m = fmaxf(fmaxf(fmaxf(c0.x, c0.y), fmaxf(c0.z, c0.w)),
              fmaxf(fmaxf(fmaxf(c1.x, c1.y), fmaxf(c1.z, c1.w)),
              fmaxf(fmaxf(fmaxf(c2.x, c2.y), fmaxf(c2.z, c2.w)),
                    fmaxf(fmaxf(c3.x, c3.y), fmaxf(c3.z, c3.w)))));
#pragma unroll
    for (int o = 16; o > 0; o >>= 1) m = fmaxf(m, __shfl_xor(m, o, 32));

    float s = __expf(c0.x - m) + __expf(c0.y - m) + __expf(c0.z - m) + __expf(c0.w - m)
            + __expf(c1.x - m) + __expf(c1.y - m) + __expf(c1.z - m) + __expf(c1.w - m)
            + __expf(c2.x - m) + __expf(c2.y - m) + __expf(c2.z - m) + __expf(c2.w - m)
            + __expf(c3.x - m) + __expf(c3.y - m) + __expf(c3.z - m) + __expf(c3.w - m);
#pragma unroll
    for (int o = 16; o > 0; o >>= 1) s += __shfl_xor(s, o, 32);

    const float lse = m + __logf(s);

    if (lane == 0) {
      const int b   = r / (Tn * (Un + 1));
      const int rem = r - b * (Tn * (Un + 1));
      const int t   = rem / (Un + 1);
      const int u   = rem - t * (Un + 1);
      blank_lp[r] = buf[BLANKC] - lse;      // random LDS gather, buffer still live
      if (u < Un) {
        const int tok = tokens[b * Un + u];
        emit_lp[(b * Tn + t) * Un + u] = buf[tok] - lse;
      }
    }

    if (!has_next) break;
    r = rn;
    bufi ^= 1;
  }
}

// ---------------------------------------------------------------------------
// Phase 2: forward DP via anti-diagonal wavefront. One block per batch item.
// alpha[t][u] = logaddexp(alpha[t-1][u]+blank[t-1][u], alpha[t][u-1]+emit[t][u-1])
// ---------------------------------------------------------------------------
__global__ __launch_bounds__(128) void rnnt_dp_kernel(
    const float* __restrict__ blank_lp,
    const float* __restrict__ emit_lp,
    const int*   __restrict__ src_len,
    const int*   __restrict__ tokens_len,
    const int*   __restrict__ mel_len,
    float*       __restrict__ partial)    // (B,) loss/mel_len
{
  const int b   = blockIdx.x;
  const int tid = threadIdx.x;

  __shared__ float bufA[Un + 1];
  __shared__ float bufB[Un + 1];
  float* prev = bufA;
  float* cur  = bufB;

  const int   t_last  = src_len[b] - 1;
  const int   u_last  = tokens_len[b];
  const float inv_mel = 1.0f / (float)mel_len[b];
  const float* Bl = blank_lp + (size_t)b * Tn * (Un + 1);
  const float* Em = emit_lp  + (size_t)b * Tn * Un;

  for (int d = 0; d < Tn + Un; ++d) {
    const int u = tid;
    if (u <= Un && u <= d) {
      const int t = d - u;
      if (t < Tn) {
        float val;
        if (d == 0) {
          val = 0.0f;                                   // alpha[0][0]
        } else {
          const float fb = (t >= 1) ? prev[u]     + Bl[(t - 1) * (Un + 1) + u] : NEG_INF_F;
          const float fe = (u >= 1) ? prev[u - 1] + Em[t * Un + (u - 1)]       : NEG_INF_F;
          val = logaddexp_f(fb, fe);
        }
        cur[u] = val;
        if (t == t_last && u == u_last) {
          partial[b] = -(val + Bl[t_last * (Un + 1) + u_last]) * inv_mel;
        }
      }
    }
    __syncthreads();
    float* tmp = prev; prev = cur; cur = tmp;
  }
}

// ---------------------------------------------------------------------------
// Phase 3: mean over batch -> scalar f32 output.
// ---------------------------------------------------------------------------
__global__ void rnnt_finalize_kernel(const float* __restrict__ partial,
                                     float* __restrict__ out)
{
  if (threadIdx.x == 0 && blockIdx.x == 0) {
    float s = 0.0f;
#pragma unroll
    for (int i = 0; i < Bn; ++i) s += partial[i];
    out[0] = s * (1.0f / (float)Bn);
  }
}

extern "C" void kernel_launch(void* const* d_in, const int* in_sizes, int n_in,
                              void* d_out, int out_size, void* d_ws, size_t ws_size,
                              hipStream_t stream)
{
  (void)in_sizes; (void)n_in; (void)out_size; (void)ws_size;

  const float* logits     = (const float*)d_in[0];   // (B,T,U+1,V) f32
  const int*   tokens     = (const int*)  d_in[1];   // (B,U)
  const int*   src_len    = (const int*)  d_in[2];   // (B,)
  const int*   tokens_len = (const int*)  d_in[3];   // (B,)
  const int*   mel_len    = (const int*)  d_in[4];   // (B,)

  float* blank_lp = (float*)d_ws;                    // NROWS floats
  float* emit_lp  = blank_lp + NROWS;                // NEMIT floats
  float* partial  = emit_lp + NEMIT;                 // Bn floats
  // total workspace use: ~2.57 MB

  rnnt_lse_kernel<<<dim3(2048), dim3(256), 0, stream>>>(logits, tokens,
                                                        blank_lp, emit_lp);
  rnnt_dp_kernel<<<dim3(Bn), dim3(128), 0, stream>>>(blank_lp, emit_lp,
                                                     src_len, tokens_len,
                                                     mel_len, partial);
  rnnt_finalize_kernel<<<dim3(1), dim3(32), 0, stream>>>(partial, (float*)d_out);
}